// GAT_27771258536144
// MI455X (gfx1250) — compile-verified
//
#include <hip/hip_runtime.h>
#include <math.h>

// ---------------------------------------------------------------------------
// GAT (2-layer) + global mean pool for MI455X (gfx1250, wave32).
// fp32 end-to-end (matches reference numerics). GEMM1 via V_WMMA_F32_16X16X4_F32.
// Edge passes are per-(edge,channel) threaded for coalesced gather/scatter.
// ---------------------------------------------------------------------------

typedef float v2f __attribute__((ext_vector_type(2)));
typedef float v8f __attribute__((ext_vector_type(8)));

#define FIN 128
#define H1  8
#define C1  8
#define O1  64      // H1*C1
#define O2  8
#define NB  64      // graphs in batch

// ---- float <-> order-preserving uint (for atomicMax on floats) ------------
static __device__ __forceinline__ unsigned f2ord(float f) {
  unsigned u = __float_as_uint(f);
  return (u & 0x80000000u) ? ~u : (u | 0x80000000u);
}
static __device__ __forceinline__ float ord2f(unsigned u) {
  return (u & 0x80000000u) ? __uint_as_float(u & 0x7fffffffu)
                           : __uint_as_float(~u);
}
static __device__ __forceinline__ float lrelu(float a) {
  return a > 0.f ? a : 0.2f * a;
}
static __device__ __forceinline__ void edge_sd(const int* __restrict__ ei,
                                               long long e, long long E,
                                               int& s, int& d) {
  if (e < E) { s = ei[e]; d = ei[E + e]; }
  else       { s = d = (int)(e - E); }          // self loops appended
}

// ---- fill ------------------------------------------------------------------
__global__ void k_fill_u32(unsigned* __restrict__ p, long long n, unsigned v) {
  long long i = (long long)blockIdx.x * 256 + threadIdx.x;
  long long stride = (long long)gridDim.x * 256;
  for (; i < n; i += stride) p[i] = v;
}

// ---- GEMM1: xh1 = x @ W1 using fp32 WMMA ----------------------------------
// block = 128 threads = 4 waves; block covers 16 rows x 64 cols (wave = 16x16).
__global__ void k_gemm1_wmma(const float* __restrict__ x,
                             const float* __restrict__ W,
                             float* __restrict__ xh, int n_nodes) {
  const int lane = threadIdx.x & 31;
  const int wave = threadIdx.x >> 5;            // 0..3 -> column tile
  const int m0   = blockIdx.x * 16;
  const int n0   = wave * 16;
  const int lo   = lane & 15;                   // M row (A) / N col (B,D)
  const int kg   = lane >> 4;                   // K group (0 or 1)
  int mrow = m0 + lo; if (mrow >= n_nodes) mrow = n_nodes - 1;
  const float* __restrict__ xr = x + (size_t)mrow * FIN;

  v8f c = {0.f, 0.f, 0.f, 0.f, 0.f, 0.f, 0.f, 0.f};
  #pragma unroll
  for (int k = 0; k < FIN; k += 4) {
    v2f a, b;
    a.x = xr[k + 2 * kg + 0];                   // A[m][k + 2*kg + {0,1}]
    a.y = xr[k + 2 * kg + 1];
    b.x = W[(size_t)(k + 2 * kg + 0) * O1 + n0 + lo];  // B[k + 2*kg + {0,1}][n]
    b.y = W[(size_t)(k + 2 * kg + 1) * O1 + n0 + lo];
    c = __builtin_amdgcn_wmma_f32_16x16x4_f32(false, a, false, b,
                                              (short)0, c, false, false);
  }
  #pragma unroll
  for (int r = 0; r < 8; ++r) {                 // VGPR r: M=r (lanes 0-15), M=r+8 (16-31)
    int row = m0 + r + kg * 8;
    if (row < n_nodes) xh[(size_t)row * O1 + n0 + lo] = c[r];
  }
}

// ---- attention logits layer 1: a1s/a1d[n,h] = sum_c xh[n,h,c]*att[h,c] ----
__global__ void k_att1(const float* __restrict__ xh,
                       const float* __restrict__ asw, const float* __restrict__ adw,
                       float* __restrict__ a1s, float* __restrict__ a1d,
                       long long n_nodes) {
  long long t = (long long)blockIdx.x * 256 + threadIdx.x;   // n*8 + h
  if (t >= n_nodes * H1) return;
  int h = (int)(t & 7);
  const float* row = xh + (t >> 3) * O1 + h * C1;
  float s = 0.f, d = 0.f;
  #pragma unroll
  for (int c = 0; c < C1; ++c) {
    s += row[c] * asw[h * C1 + c];
    d += row[c] * adw[h * C1 + c];
  }
  a1s[t] = s; a1d[t] = d;
}

// ---- edge pass 1a: segment max ---------------------------------------------
__global__ void k_emax1(const int* __restrict__ ei, long long E, long long Etot,
                        const float* __restrict__ a1s, const float* __restrict__ a1d,
                        unsigned* __restrict__ amax) {
  long long e = (long long)blockIdx.x * 256 + threadIdx.x;
  if (e >= Etot) return;
  int s, d; edge_sd(ei, e, E, s, d);
  #pragma unroll
  for (int h = 0; h < H1; ++h) {
    float a = lrelu(a1s[(size_t)s * H1 + h] + a1d[(size_t)d * H1 + h]);
    atomicMax(&amax[(size_t)d * H1 + h], f2ord(a));
  }
}

// ---- edge pass 1b: exp + segment sum ---------------------------------------
__global__ void k_esum1(const int* __restrict__ ei, long long E, long long Etot,
                        const float* __restrict__ a1s, const float* __restrict__ a1d,
                        const unsigned* __restrict__ amax, float* __restrict__ den) {
  long long e = (long long)blockIdx.x * 256 + threadIdx.x;
  if (e >= Etot) return;
  int s, d; edge_sd(ei, e, E, s, d);
  #pragma unroll
  for (int h = 0; h < H1; ++h) {
    float a = lrelu(a1s[(size_t)s * H1 + h] + a1d[(size_t)d * H1 + h]);
    float ex = __expf(a - ord2f(amax[(size_t)d * H1 + h]));
    atomicAdd(&den[(size_t)d * H1 + h], ex);
  }
}

// ---- edge pass 1c: weighted message scatter (thread = edge*64 + channel) ---
__global__ void k_emsg1(const int* __restrict__ ei, long long E, long long Etot,
                        const float* __restrict__ a1s, const float* __restrict__ a1d,
                        const unsigned* __restrict__ amax, const float* __restrict__ den,
                        const float* __restrict__ xh, float* __restrict__ agg) {
  long long t = (long long)blockIdx.x * 256 + threadIdx.x;
  if (t >= Etot * O1) return;
  long long e = t >> 6;
  int j = (int)(t & 63);
  int h = j >> 3;
  int s, d; edge_sd(ei, e, E, s, d);
  float a = lrelu(a1s[(size_t)s * H1 + h] + a1d[(size_t)d * H1 + h]);
  float w = __expf(a - ord2f(amax[(size_t)d * H1 + h])) /
            (den[(size_t)d * H1 + h] + 1e-16f);
  atomicAdd(&agg[(size_t)d * O1 + j], xh[(size_t)s * O1 + j] * w);
}

// ---- finalize layer 1: elu(agg + b1) in place ------------------------------
__global__ void k_fin1(float* __restrict__ agg, const float* __restrict__ b1,
                       long long n) {
  long long t = (long long)blockIdx.x * 256 + threadIdx.x;
  if (t >= n) return;
  float v = agg[t] + b1[t & 63];
  agg[t] = v > 0.f ? v : expm1f(v);
}

// ---- GEMM2 + attention logits layer 2 (W2 staged in LDS) -------------------
__global__ void k_gemm2(const float* __restrict__ h, const float* __restrict__ W2,
                        const float* __restrict__ asw, const float* __restrict__ adw,
                        float* __restrict__ xh2, float* __restrict__ a2s,
                        float* __restrict__ a2d, long long n_nodes) {
  __shared__ float w[O1 * O2];
  for (int i = threadIdx.x; i < O1 * O2; i += blockDim.x) w[i] = W2[i];
  __syncthreads();
  long long n = (long long)blockIdx.x * 256 + threadIdx.x;
  if (n >= n_nodes) return;
  const float* row = h + n * O1;
  float acc[O2];
  #pragma unroll
  for (int c = 0; c < O2; ++c) acc[c] = 0.f;
  for (int k = 0; k < O1; ++k) {
    float hv = row[k];
    #pragma unroll
    for (int c = 0; c < O2; ++c) acc[c] += hv * w[k * O2 + c];
  }
  float s = 0.f, d = 0.f;
  #pragma unroll
  for (int c = 0; c < O2; ++c) {
    xh2[n * O2 + c] = acc[c];
    s += acc[c] * asw[c];
    d += acc[c] * adw[c];
  }
  a2s[n] = s; a2d[n] = d;
}

// ---- edge passes layer 2 (single head) -------------------------------------
__global__ void k_emax2(const int* __restrict__ ei, long long E, long long Etot,
                        const float* __restrict__ a2s, const float* __restrict__ a2d,
                        unsigned* __restrict__ amax) {
  long long e = (long long)blockIdx.x * 256 + threadIdx.x;
  if (e >= Etot) return;
  int s, d; edge_sd(ei, e, E, s, d);
  atomicMax(&amax[d], f2ord(lrelu(a2s[s] + a2d[d])));
}

__global__ void k_esum2(const int* __restrict__ ei, long long E, long long Etot,
                        const float* __restrict__ a2s, const float* __restrict__ a2d,
                        const unsigned* __restrict__ amax, float* __restrict__ den) {
  long long e = (long long)blockIdx.x * 256 + threadIdx.x;
  if (e >= Etot) return;
  int s, d; edge_sd(ei, e, E, s, d);
  float a = lrelu(a2s[s] + a2d[d]);
  atomicAdd(&den[d], __expf(a - ord2f(amax[d])));
}

__global__ void k_emsg2(const int* __restrict__ ei, long long E, long long Etot,
                        const float* __restrict__ a2s, const float* __restrict__ a2d,
                        const unsigned* __restrict__ amax, const float* __restrict__ den,
                        const float* __restrict__ xh2, float* __restrict__ agg) {
  long long t = (long long)blockIdx.x * 256 + threadIdx.x;
  if (t >= Etot * O2) return;
  long long e = t >> 3;
  int c = (int)(t & 7);
  int s, d; edge_sd(ei, e, E, s, d);
  float a = lrelu(a2s[s] + a2d[d]);
  float w = __expf(a - ord2f(amax[d])) / (den[d] + 1e-16f);
  atomicAdd(&agg[(size_t)d * O2 + c], xh2[(size_t)s * O2 + c] * w);
}

// ---- pooling: per-block LDS bins (batch is sorted -> intra-block hits LDS) -
__global__ void k_pool(const float* __restrict__ agg2, const int* __restrict__ batch,
                       float* __restrict__ sums, float* __restrict__ cnt,
                       long long n_nodes) {
  __shared__ float ls[NB * O2];
  __shared__ float lc[NB];
  for (int i = threadIdx.x; i < NB * O2; i += blockDim.x) ls[i] = 0.f;
  for (int i = threadIdx.x; i < NB; i += blockDim.x) lc[i] = 0.f;
  __syncthreads();
  long long n = (long long)blockIdx.x * 256 + threadIdx.x;
  if (n < n_nodes) {
    int b = batch[n];
    #pragma unroll
    for (int c = 0; c < O2; ++c) atomicAdd(&ls[b * O2 + c], agg2[n * O2 + c]);
    atomicAdd(&lc[b], 1.f);
  }
  __syncthreads();
  for (int i = threadIdx.x; i < NB * O2; i += blockDim.x) atomicAdd(&sums[i], ls[i]);
  for (int i = threadIdx.x; i < NB; i += blockDim.x) atomicAdd(&cnt[i], lc[i]);
}

__global__ void k_final(const float* __restrict__ sums, const float* __restrict__ cnt,
                        const float* __restrict__ b2, float* __restrict__ out) {
  int t = blockIdx.x * 256 + threadIdx.x;
  if (t >= NB * O2) return;
  out[t] = sums[t] / fmaxf(cnt[t >> 3], 1.f) + b2[t & 7];
}

// ---------------------------------------------------------------------------
extern "C" void kernel_launch(void* const* d_in, const int* in_sizes, int n_in,
                              void* d_out, int out_size, void* d_ws, size_t ws_size,
                              hipStream_t stream) {
  const float* x    = (const float*)d_in[0];
  const int*   ei   = (const int*)d_in[1];
  const int*   batch= (const int*)d_in[2];
  const float* W1   = (const float*)d_in[3];
  const float* a1sw = (const float*)d_in[4];
  const float* a1dw = (const float*)d_in[5];
  const float* b1   = (const float*)d_in[6];
  const float* W2   = (const float*)d_in[7];
  const float* a2sw = (const float*)d_in[8];
  const float* a2dw = (const float*)d_in[9];
  const float* b2   = (const float*)d_in[10];
  float* out = (float*)d_out;

  const long long N    = in_sizes[0] / FIN;
  const long long E    = in_sizes[1] / 2;
  const long long Etot = E + N;                 // edges + self loops

  // workspace layout (fp32 words)
  float* ws = (float*)d_ws;
  size_t off = 0;
  float*    xh1   = ws + off; off += (size_t)N * O1;
  float*    a1s   = ws + off; off += (size_t)N * H1;
  float*    a1d   = ws + off; off += (size_t)N * H1;
  unsigned* amax1 = (unsigned*)(ws + off); off += (size_t)N * H1;
  float*    den1  = ws + off; off += (size_t)N * H1;
  float*    agg1  = ws + off; off += (size_t)N * O1;   // layer1 out -> h (elu'd)
  float*    xh2   = ws + off; off += (size_t)N * O2;
  float*    a2s   = ws + off; off += (size_t)N;
  float*    a2d   = ws + off; off += (size_t)N;
  unsigned* amax2 = (unsigned*)(ws + off); off += (size_t)N;
  float*    den2  = ws + off; off += (size_t)N;
  float*    agg2  = ws + off; off += (size_t)N * O2;
  float*    sums  = ws + off; off += NB * O2;
  float*    cnt   = ws + off; off += NB;
  (void)ws_size; (void)n_in; (void)out_size;

  auto nb = [](long long n, int t) { return (unsigned)((n + t - 1) / t); };
  const unsigned ORDNEG = 0x00800000u;          // f2ord(-FLT_MAX)

  // init accumulators (graph is replayed -> must re-init every call)
  k_fill_u32<<<nb(N * H1, 256), 256, 0, stream>>>(amax1, N * H1, ORDNEG);
  k_fill_u32<<<nb(N * H1, 256), 256, 0, stream>>>((unsigned*)den1, N * H1, 0u);
  k_fill_u32<<<nb(N * O1, 256), 256, 0, stream>>>((unsigned*)agg1, N * O1, 0u);
  k_fill_u32<<<nb(N, 256), 256, 0, stream>>>(amax2, N, ORDNEG);
  k_fill_u32<<<nb(N, 256), 256, 0, stream>>>((unsigned*)den2, N, 0u);
  k_fill_u32<<<nb(N * O2, 256), 256, 0, stream>>>((unsigned*)agg2, N * O2, 0u);
  k_fill_u32<<<nb(NB * O2 + NB, 256), 256, 0, stream>>>((unsigned*)sums,
                                                        NB * O2 + NB, 0u);

  // layer 1
  k_gemm1_wmma<<<nb(N, 16), 128, 0, stream>>>(x, W1, xh1, (int)N);
  k_att1<<<nb(N * H1, 256), 256, 0, stream>>>(xh1, a1sw, a1dw, a1s, a1d, N);
  k_emax1<<<nb(Etot, 256), 256, 0, stream>>>(ei, E, Etot, a1s, a1d, amax1);
  k_esum1<<<nb(Etot, 256), 256, 0, stream>>>(ei, E, Etot, a1s, a1d, amax1, den1);
  k_emsg1<<<nb(Etot * O1, 256), 256, 0, stream>>>(ei, E, Etot, a1s, a1d, amax1,
                                                  den1, xh1, agg1);
  k_fin1<<<nb(N * O1, 256), 256, 0, stream>>>(agg1, b1, N * O1);

  // layer 2
  k_gemm2<<<nb(N, 256), 256, 0, stream>>>(agg1, W2, a2sw, a2dw, xh2, a2s, a2d, N);
  k_emax2<<<nb(Etot, 256), 256, 0, stream>>>(ei, E, Etot, a2s, a2d, amax2);
  k_esum2<<<nb(Etot, 256), 256, 0, stream>>>(ei, E, Etot, a2s, a2d, amax2, den2);
  k_emsg2<<<nb(Etot * O2, 256), 256, 0, stream>>>(ei, E, Etot, a2s, a2d, amax2,
                                                  den2, xh2, agg2);

  // global mean pool
  k_pool<<<nb(N, 256), 256, 0, stream>>>(agg2, batch, sums, cnt, N);
  k_final<<<nb(NB * O2, 256), 256, 0, stream>>>(sums, cnt, b2, out);
}